// QuanvolutionSelfAttentionClassifier_65481071401740
// MI455X (gfx1250) — compile-verified
//
#include <hip/hip_runtime.h>
#include <math.h>

typedef float v2f __attribute__((ext_vector_type(2)));
typedef float v8f __attribute__((ext_vector_type(8)));

struct Cplx { float re, im; };
__device__ inline Cplx cmul(Cplx a, Cplx b){ return {a.re*b.re - a.im*b.im, a.re*b.im + a.im*b.re}; }
__device__ inline Cplx cadd(Cplx a, Cplx b){ return {a.re+b.re, a.im+b.im}; }

// ---------------- Kernel 0: tiny 4-qubit circuit -> attn[4] ----------------
__global__ void attn_kernel(const float* __restrict__ rot,
                            const float* __restrict__ ent,
                            float* __restrict__ attn_out) {
  if (threadIdx.x != 0 || blockIdx.x != 0) return;
  Cplx psi[16];
  for (int i = 0; i < 16; ++i) psi[i] = {0.f, 0.f};
  psi[0] = {1.f, 0.f};
  // single-qubit gates: g = Rz * Ry * Rx on qubit q (bit q of index)
  for (int q = 0; q < 4; ++q) {
    float cx = __cosf(rot[3*q+0]*0.5f), sx = __sinf(rot[3*q+0]*0.5f);
    float cy = __cosf(rot[3*q+1]*0.5f), sy = __sinf(rot[3*q+1]*0.5f);
    float cz = __cosf(rot[3*q+2]*0.5f), sz = __sinf(rot[3*q+2]*0.5f);
    Cplx rx[2][2] = {{{cx,0.f},{0.f,-sx}},{{0.f,-sx},{cx,0.f}}};
    Cplx ry[2][2] = {{{cy,0.f},{-sy,0.f}},{{sy,0.f},{cy,0.f}}};
    Cplx rz[2][2] = {{{cz,-sz},{0.f,0.f}},{{0.f,0.f},{cz,sz}}};
    Cplx m[2][2], g[2][2];
    for (int r = 0; r < 2; ++r)
      for (int c = 0; c < 2; ++c)
        m[r][c] = cadd(cmul(ry[r][0], rx[0][c]), cmul(ry[r][1], rx[1][c]));
    for (int r = 0; r < 2; ++r)
      for (int c = 0; c < 2; ++c)
        g[r][c] = cadd(cmul(rz[r][0], m[0][c]), cmul(rz[r][1], m[1][c]));
    int bit = 1 << q;
    for (int s = 0; s < 16; ++s) {
      if (s & bit) continue;
      Cplx a = psi[s], b = psi[s | bit];
      psi[s]       = cadd(cmul(g[0][0], a), cmul(g[0][1], b));
      psi[s | bit] = cadd(cmul(g[1][0], a), cmul(g[1][1], b));
    }
  }
  // controlled-RX: if bit q set, apply Rx(ent[q]) on bit q+1
  for (int q = 0; q < 3; ++q) {
    float c = __cosf(ent[q]*0.5f), s = __sinf(ent[q]*0.5f);
    Cplx g[2][2] = {{{c,0.f},{0.f,-s}},{{0.f,-s},{c,0.f}}};
    int cb = 1 << q, tb = 1 << (q + 1);
    for (int st = 0; st < 16; ++st) {
      if (!(st & cb) || (st & tb)) continue;
      Cplx a = psi[st], b = psi[st | tb];
      psi[st]      = cadd(cmul(g[0][0], a), cmul(g[0][1], b));
      psi[st | tb] = cadd(cmul(g[1][0], a), cmul(g[1][1], b));
    }
  }
  float probs[16];
  for (int i = 0; i < 16; ++i) probs[i] = psi[i].re*psi[i].re + psi[i].im*psi[i].im;
  for (int q = 0; q < 4; ++q) {
    float a = 0.f;
    for (int j = 0; j < 16; ++j) a += (1.f - 2.f*((j >> q) & 1)) * probs[j];
    attn_out[q] = a;
  }
}

// ------------- Kernel 1: V[p][n][c] = sum_i sign(n,i)*attn[i]*W[c,4p+i] -------------
__global__ void buildV_kernel(const float* __restrict__ W,
                              const float* __restrict__ attn,
                              float* __restrict__ V) {
  int t = blockIdx.x * blockDim.x + threadIdx.x;
  if (t >= 196 * 16) return;
  int p = t >> 4, n = t & 15;
  float sg[4];
  for (int i = 0; i < 4; ++i)
    sg[i] = (1.f - 2.f * ((n >> (3 - i)) & 1)) * attn[i];
  for (int c = 0; c < 10; ++c) {
    float v = 0.f;
    for (int i = 0; i < 4; ++i) v += sg[i] * W[c * 784 + 4 * p + i];
    V[t * 10 + c] = v;
  }
}

// ------------- Kernel 2: fused quanv (WMMA) + classifier + log_softmax -------------
__global__ __launch_bounds__(256) void quanv_kernel(
    const float* __restrict__ x, const float* __restrict__ U,
    const float* __restrict__ V, const float* __restrict__ bias,
    float* __restrict__ out) {
  __shared__ float wp[8][16];
  const int b    = blockIdx.x;            // sample
  const int lane = threadIdx.x & 31;
  const int wave = threadIdx.x >> 5;
  const int half = lane >> 4;             // K-half select for A/B fragments
  const int l16  = lane & 15;

  // B fragments: B[k][n] = U^T[k][n] = U[n][k]; lane holds column n = l16.
  // k = 4t + 2*half is even -> 8B-aligned float2 load.
  v2f bfrag[4];
#pragma unroll
  for (int t = 0; t < 4; ++t)
    bfrag[t] = *(const v2f*)(U + l16 * 16 + 4 * t + 2 * half);

  float acc[10];
#pragma unroll
  for (int c = 0; c < 10; ++c) acc[c] = 0.f;

  const float* xbase = x + (size_t)b * 784;

  for (int tile = wave; tile < 13; tile += 8) {     // 13 tiles of 16 patches (196 valid)
    int p_mine = tile * 16 + l16;                   // A-matrix row for this lane
    float valid = (p_mine < 196) ? 1.f : 0.f;       // pad rows -> psi == 0
    int pc = (p_mine < 196) ? p_mine : 195;         // clamped for addressing
    int r14 = pc / 14, c14 = pc % 14;
    const float* xb = xbase + (size_t)(2 * r14) * 28 + 2 * c14;
    v2f top = *(const v2f*)(xb);                    // pixels k=0,1 (row 2r14)
    v2f bot = *(const v2f*)(xb + 28);               // pixels k=2,3 (row 2r14+1)

    // prefetch next tile's patch for this lane (global_prefetch_b8)
    int p_next = (tile + 8) * 16 + l16;
    if (p_next < 196) {
      int rn = p_next / 14, cn = p_next % 14;
      __builtin_prefetch(xbase + (size_t)(2 * rn) * 28 + 2 * cn, 0, 0);
    }

    float c0 = __cosf(top.x * 0.5f), s0 = __sinf(top.x * 0.5f);
    float c1 = __cosf(top.y * 0.5f), s1 = __sinf(top.y * 0.5f);
    float c2 = __cosf(bot.x * 0.5f), s2 = __sinf(bot.x * 0.5f);
    float c3 = __cosf(bot.y * 0.5f), s3 = __sinf(bot.y * 0.5f);

    // psi[m] = q0[b3]*q1[b2]*q2[b1]*q3[b0], m = 4t + 2*half + j:
    //   b3 = t>>1, b2 = t&1, b1 = half, b0 = j
    float e2 = (half ? s2 : c2) * valid;            // folds validity -> zero rows
    float base0 = c0 * c1 * e2;                     // t=0
    float base1 = c0 * s1 * e2;                     // t=1
    float base2 = s0 * c1 * e2;                     // t=2
    float base3 = s0 * s1 * e2;                     // t=3

    v2f a0, a1, a2, a3;
    a0.x = base0 * c3; a0.y = base0 * s3;
    a1.x = base1 * c3; a1.y = base1 * s3;
    a2.x = base2 * c3; a2.y = base2 * s3;
    a3.x = base3 * c3; a3.y = base3 * s3;

    // psi_out(16x16) = psi(16x16) @ U^T via 4 chained f32 WMMAs (K=4 each)
    v8f d = {0.f, 0.f, 0.f, 0.f, 0.f, 0.f, 0.f, 0.f};
    d = __builtin_amdgcn_wmma_f32_16x16x4_f32(false, a0, false, bfrag[0], (short)0, d, false, false);
    d = __builtin_amdgcn_wmma_f32_16x16x4_f32(false, a1, false, bfrag[1], (short)0, d, false, false);
    d = __builtin_amdgcn_wmma_f32_16x16x4_f32(false, a2, false, bfrag[2], (short)0, d, false, false);
    d = __builtin_amdgcn_wmma_f32_16x16x4_f32(false, a3, false, bfrag[3], (short)0, d, false, false);

    // D layout: vgpr r -> row (r + 8*half), col n = l16. probs = d*d; then
    // logits partials against the folded table V[p][n][c] (hot in L2, 125 KB).
#pragma unroll
    for (int r = 0; r < 8; ++r) {
      int p = tile * 16 + r + 8 * half;
      if (p > 195) p = 195;                          // safe: pr==0 for pad rows
      float pr = d[r] * d[r];
      const float* vp = V + ((size_t)p * 16 + l16) * 10;
#pragma unroll
      for (int c = 0; c < 10; ++c) acc[c] = fmaf(pr, vp[c], acc[c]);
    }
  }

  // wave32 butterfly reduce 10 partials across all lanes
#pragma unroll
  for (int c = 0; c < 10; ++c) {
    float v = acc[c];
    for (int m = 16; m >= 1; m >>= 1) v += __shfl_xor(v, m, 32);
    acc[c] = v;
  }
  if (lane == 0)
    for (int c = 0; c < 10; ++c) wp[wave][c] = acc[c];
  __syncthreads();

  if (threadIdx.x == 0) {
    float lg[10], mx = -INFINITY;
    for (int c = 0; c < 10; ++c) {
      float v = bias[c];
      for (int w = 0; w < 8; ++w) v += wp[w][c];
      lg[c] = v;
      if (v > mx) mx = v;
    }
    float se = 0.f;
    for (int c = 0; c < 10; ++c) se += __expf(lg[c] - mx);
    float lse = __logf(se);
    for (int c = 0; c < 10; ++c) out[(size_t)b * 10 + c] = lg[c] - mx - lse;
  }
}

extern "C" void kernel_launch(void* const* d_in, const int* in_sizes, int n_in,
                              void* d_out, int out_size, void* d_ws, size_t ws_size,
                              hipStream_t stream) {
  const float* x    = (const float*)d_in[0];  // (B, 784)
  const float* Uq   = (const float*)d_in[1];  // (16, 16)
  const float* rot  = (const float*)d_in[2];  // (12,)
  const float* ent  = (const float*)d_in[3];  // (3,)
  const float* W    = (const float*)d_in[4];  // (10, 784)
  const float* bias = (const float*)d_in[5];  // (10,)
  float* out = (float*)d_out;                 // (B, 10)
  const int B = in_sizes[0] / 784;

  float* ws_attn = (float*)d_ws;              // 4 floats
  float* ws_V    = (float*)d_ws + 16;         // 196*16*10 floats (~125 KB)

  attn_kernel<<<1, 1, 0, stream>>>(rot, ent, ws_attn);
  buildV_kernel<<<(196 * 16 + 255) / 256, 256, 0, stream>>>(W, ws_attn, ws_V);
  quanv_kernel<<<B, 256, 0, stream>>>(x, Uq, ws_V, bias, out);
}